// MutualMultiAttentionHead_90357521973751
// MI455X (gfx1250) — compile-verified
//
#include <hip/hip_runtime.h>
#include <hip/hip_bf16.h>

// Problem constants (from the reference)
#define GN   4096   // graph nodes
#define GE   4096   // graph edges (= line-graph nodes)
#define GELG 16384  // line-graph edges
#define GH   4      // heads
#define GFN  64     // node feats
#define GFE  32     // edge feats

typedef __attribute__((ext_vector_type(2))) float v2f;
typedef __attribute__((ext_vector_type(8))) float v8f;

// ---------------------------------------------------------------------------
// Generic fp32 WMMA GEMM:  C[M,Nc] = A[M,K] @ B[K,Nc] (+ bias[Nc])
// Requirements: M % 16 == 0, K % 4 == 0. Nc arbitrary (cols >= Nc masked).
// One wave per 16x16 output tile, 4 waves per block.
// A fragment (16x4 f32): lanes 0-15 = rows M0..M15, hl=lane>>4 selects K-half,
//   vgpr j holds K = k + 2*hl + j  -> one global_load_b64 per lane per wmma.
// B fragment (4x16): lane&15 = column, vgpr j holds K = k + 2*hl + j.
// C/D (16x16 f32, 8 VGPRs): lane&15 = column, vgpr i -> row i + 8*(lane>>4).
// ---------------------------------------------------------------------------
__global__ void gemm_f32_wmma(const float* __restrict__ A,
                              const float* __restrict__ B,
                              const float* __restrict__ bias,
                              float* __restrict__ C,
                              int M, int K, int Nc) {
  const int nTiles = (Nc + 15) >> 4;
  const int totalTiles = (M >> 4) * nTiles;
  const int tile = blockIdx.x * 4 + (threadIdx.x >> 5);
  if (tile >= totalTiles) return;           // uniform per wave; EXEC stays all-1s
  const int tm = tile / nTiles;
  const int tn = tile - tm * nTiles;
  const int m0 = tm << 4, n0 = tn << 4;
  const int lane = threadIdx.x & 31;
  const int ml = lane & 15;                 // A: row in tile; B/C: col in tile
  const int hl = lane >> 4;                 // K-half selector / M-half for C
  const int colG = n0 + ml;
  const float bmask = (colG < Nc) ? 1.0f : 0.0f;   // zero out padded columns
  const int colS = (colG < Nc) ? colG : (Nc - 1);  // keep loads in-bounds
  const float* __restrict__ Ap = A + (size_t)(m0 + ml) * (size_t)K + (hl << 1);
  const float* __restrict__ Bp = B + (size_t)(hl << 1) * (size_t)Nc + colS;

  v8f acc = {};
#pragma unroll 4
  for (int k = 0; k < K; k += 4) {
    v2f a = *(const v2f*)(Ap + k);                 // A[m][k+2hl], A[m][k+2hl+1]
    v2f b;
    b.x = Bp[(size_t)k * Nc] * bmask;              // B[k+2hl  ][col]
    b.y = Bp[(size_t)k * Nc + Nc] * bmask;         // B[k+2hl+1][col]
    acc = __builtin_amdgcn_wmma_f32_16x16x4_f32(
        /*neg_a=*/false, a, /*neg_b=*/false, b,
        /*c_mod=*/(short)0, acc, /*reuse_a=*/false, /*reuse_b=*/false);
  }

  if (colG < Nc) {
    const float bb = bias ? bias[colG] : 0.0f;
#pragma unroll
    for (int i = 0; i < 8; ++i) {
      C[(size_t)(m0 + i + (hl << 3)) * Nc + colG] = acc[i] + bb;
    }
  }
}

// out[n] = b2[n] + sum_k b1[k] * W2[k,n]   (bias fold of the collapsed MLP)
__global__ void fold_bias(const float* __restrict__ b1, const float* __restrict__ W2,
                          const float* __restrict__ b2, float* __restrict__ out,
                          int K, int Nc) {
  int n = blockIdx.x * blockDim.x + threadIdx.x;
  if (n >= Nc) return;
  float s = b2[n];
  for (int k = 0; k < K; ++k) s += b1[k] * W2[(size_t)k * Nc + n];
  out[n] = s;
}

// Per-head softmax over n contiguous elements: y = softmax(x * scale)
__global__ void softmax_head(const float* __restrict__ x, float* __restrict__ y,
                             int n, float scale) {
  const float* xx = x + (size_t)blockIdx.x * n;
  float* yy = y + (size_t)blockIdx.x * n;
  __shared__ float red[256];
  const int tid = threadIdx.x;

  float lmax = -3.402823466e38f;
  for (int i = tid; i < n; i += 256) lmax = fmaxf(lmax, xx[i] * scale);
  red[tid] = lmax; __syncthreads();
  for (int s = 128; s > 0; s >>= 1) {
    if (tid < s) red[tid] = fmaxf(red[tid], red[tid + s]);
    __syncthreads();
  }
  const float m = red[0]; __syncthreads();

  float lsum = 0.0f;
  for (int i = tid; i < n; i += 256) lsum += __expf(xx[i] * scale - m);
  red[tid] = lsum; __syncthreads();
  for (int s = 128; s > 0; s >>= 1) {
    if (tid < s) red[tid] += red[tid + s];
    __syncthreads();
  }
  const float inv = 1.0f / red[0];
  for (int i = tid; i < n; i += 256) yy[i] = __expf(xx[i] * scale - m) * inv;
}

__global__ void zero_f32(float* __restrict__ p, size_t n) {
  size_t i = (size_t)blockIdx.x * blockDim.x + threadIdx.x;
  const size_t stride = (size_t)gridDim.x * blockDim.x;
  for (; i < n; i += stride) p[i] = 0.0f;
}

// mp_n[h, src_g[e], f] += edge_attn[h,e] * nv[h, dst_g[e], f]
__global__ void scatter_node(const float* __restrict__ edge_attn,
                             const float* __restrict__ nv,
                             const int* __restrict__ src_g,
                             const int* __restrict__ dst_g,
                             float* __restrict__ mp_n) {
  int idx = blockIdx.x * blockDim.x + threadIdx.x;   // ((e*H)+h)*FN + f
  if (idx >= GE * GH * GFN) return;
  const int f = idx & (GFN - 1);
  const int eh = idx >> 6;
  const int h = eh & (GH - 1);
  const int e = eh >> 2;
  const float a = edge_attn[h * GE + e];
  const int s = src_g[e], d = dst_g[e];
  const float v = nv[(size_t)h * GN * GFN + (size_t)d * GFN + f];
  atomicAdd(&mp_n[(size_t)h * GN * GFN + (size_t)s * GFN + f], a * v);
}

// mp_e[h, src_lg[k], f] += node_attn[h, dst_g[src_lg[k]]] * ev[h, dst_lg[k], f]
__global__ void scatter_edge(const float* __restrict__ node_attn,
                             const float* __restrict__ ev,
                             const int* __restrict__ src_lg,
                             const int* __restrict__ dst_lg,
                             const int* __restrict__ dst_g,
                             float* __restrict__ mp_e) {
  int idx = blockIdx.x * blockDim.x + threadIdx.x;   // ((k*H)+h)*FE + f
  if (idx >= GELG * GH * GFE) return;
  const int f = idx & (GFE - 1);
  const int kh = idx >> 5;
  const int h = kh & (GH - 1);
  const int k = kh >> 2;
  const int s = src_lg[k], d = dst_lg[k];
  const float a = node_attn[h * GN + dst_g[s]];
  const float v = ev[(size_t)h * GE * GFE + (size_t)d * GFE + f];
  atomicAdd(&mp_e[(size_t)h * GE * GFE + (size_t)s * GFE + f], a * v);
}

// node_emb = mean_h(mp_n * nv); edge_emb = mean_h(mp_e * ev); concat into out
__global__ void finalize(const float* __restrict__ mp_n, const float* __restrict__ nv,
                         const float* __restrict__ mp_e, const float* __restrict__ ev,
                         float* __restrict__ out) {
  const int nTot = GN * GFN;   // 262144
  const int eTot = GE * GFE;   // 131072
  int i = blockIdx.x * blockDim.x + threadIdx.x;
  if (i < nTot) {
    float s = 0.0f;
#pragma unroll
    for (int h = 0; h < GH; ++h) {
      const size_t o = (size_t)h * nTot + i;
      s += mp_n[o] * nv[o];
    }
    out[i] = s * (1.0f / GH);
  } else if (i < nTot + eTot) {
    const int j = i - nTot;
    float s = 0.0f;
#pragma unroll
    for (int h = 0; h < GH; ++h) {
      const size_t o = (size_t)h * eTot + j;
      s += mp_e[o] * ev[o];
    }
    out[i] = s * (1.0f / GH);
  }
}

extern "C" void kernel_launch(void* const* d_in, const int* in_sizes, int n_in,
                              void* d_out, int out_size, void* d_ws, size_t ws_size,
                              hipStream_t stream) {
  (void)in_sizes; (void)n_in; (void)out_size; (void)ws_size;
  const float* g_adj   = (const float*)d_in[0];
  const float* lg_adj  = (const float*)d_in[1];
  const float* node_in = (const float*)d_in[2];
  const float* edge_in = (const float*)d_in[3];
  const int*   src_g   = (const int*)d_in[4];
  const int*   dst_g   = (const int*)d_in[5];
  const int*   src_lg  = (const int*)d_in[6];
  const int*   dst_lg  = (const int*)d_in[7];
  const float* W_nqk1 = (const float*)d_in[8];
  const float* b_nqk1 = (const float*)d_in[9];
  const float* W_nqk2 = (const float*)d_in[10];
  const float* b_nqk2 = (const float*)d_in[11];
  const float* W_nv1  = (const float*)d_in[12];
  const float* b_nv1  = (const float*)d_in[13];
  const float* W_nv2  = (const float*)d_in[14];
  const float* b_nv2  = (const float*)d_in[15];
  const float* W_eqk1 = (const float*)d_in[16];
  const float* b_eqk1 = (const float*)d_in[17];
  const float* W_eqk2 = (const float*)d_in[18];
  const float* b_eqk2 = (const float*)d_in[19];
  const float* W_ev1  = (const float*)d_in[20];
  const float* b_ev1  = (const float*)d_in[21];
  const float* W_ev2  = (const float*)d_in[22];
  const float* b_ev2  = (const float*)d_in[23];

  // Workspace carve-up (floats, 16B-aligned chunks). Total ~14.7 MB.
  float* ws = (float*)d_ws;
  size_t off = 0;
  auto take = [&](size_t n) { float* p = ws + off; off += (n + 3) & ~(size_t)3; return p; };
  float* AN   = take((size_t)GN * GFN);        // g_adj @ node_inputs
  float* AE   = take((size_t)GE * GFE);        // lg_adj @ edge_inputs
  float* Wnv  = take((size_t)GFN * GH * GFN);  // W_nv1 @ W_nv2   [64,256]
  float* bnv  = take(GH * GFN);
  float* Wnqk = take(GFN * GH);                // [64,4]
  float* bnqk = take(GH);
  float* Weqk = take(GFE * GH);                // [32,4]
  float* beqk = take(GH);
  float* Wev  = take((size_t)GFE * GH * GFE);  // [32,128]
  float* bev  = take(GH * GFE);
  float* nqk  = take((size_t)GN * GH);         // flat [N,H]
  float* nv   = take((size_t)GN * GH * GFN);   // flat [N,H*FN]
  float* eqk  = take((size_t)GE * GH);
  float* ev   = take((size_t)GE * GH * GFE);
  float* natt = take((size_t)GH * GN);
  float* eatt = take((size_t)GH * GE);
  float* mp_n = take((size_t)GH * GN * GFN);   // contiguous with mp_e
  float* mp_e = take((size_t)GH * GE * GFE);

  const dim3 blk(128);
  auto ggrid = [](int M, int Nc) {
    int tiles = (M >> 4) * ((Nc + 15) >> 4);
    return dim3((unsigned)((tiles + 3) / 4));
  };

  // --- Fold activation-free MLPs: (x@W1+b1)@W2+b2 == x@(W1@W2) + (b1@W2+b2) ---
  gemm_f32_wmma<<<ggrid(64, 4),   blk, 0, stream>>>(W_nqk1, W_nqk2, nullptr, Wnqk, 64, 16,   4);
  gemm_f32_wmma<<<ggrid(64, 256), blk, 0, stream>>>(W_nv1,  W_nv2,  nullptr, Wnv,  64, 1024, 256);
  gemm_f32_wmma<<<ggrid(32, 4),   blk, 0, stream>>>(W_eqk1, W_eqk2, nullptr, Weqk, 32, 16,   4);
  gemm_f32_wmma<<<ggrid(32, 128), blk, 0, stream>>>(W_ev1,  W_ev2,  nullptr, Wev,  32, 512,  128);
  fold_bias<<<1, 256, 0, stream>>>(b_nqk1, W_nqk2, b_nqk2, bnqk, 16,   4);
  fold_bias<<<1, 256, 0, stream>>>(b_nv1,  W_nv2,  b_nv2,  bnv,  1024, 256);
  fold_bias<<<1, 256, 0, stream>>>(b_eqk1, W_eqk2, b_eqk2, beqk, 16,   4);
  fold_bias<<<1, 256, 0, stream>>>(b_ev1,  W_ev2,  b_ev2,  bev,  512,  128);

  // --- Memory-bound core: stream the 64MB adjacency matrices through WMMA ---
  gemm_f32_wmma<<<ggrid(GN, GFN), blk, 0, stream>>>(g_adj,  node_in, nullptr, AN, GN, GN, GFN);
  gemm_f32_wmma<<<ggrid(GE, GFE), blk, 0, stream>>>(lg_adj, edge_in, nullptr, AE, GE, GE, GFE);

  // --- Collapsed projections ---
  gemm_f32_wmma<<<ggrid(GN, GH),       blk, 0, stream>>>(AN,      Wnqk, bnqk, nqk, GN, GFN, GH);
  gemm_f32_wmma<<<ggrid(GN, GH*GFN),   blk, 0, stream>>>(node_in, Wnv,  bnv,  nv,  GN, GFN, GH*GFN);
  gemm_f32_wmma<<<ggrid(GE, GH),       blk, 0, stream>>>(AE,      Weqk, beqk, eqk, GE, GFE, GH);
  gemm_f32_wmma<<<ggrid(GE, GH*GFE),   blk, 0, stream>>>(edge_in, Wev,  bev,  ev,  GE, GFE, GH*GFE);

  // --- Softmax over dim=1 per head (flat [N,H] buffer == reshape(H,N,1)) ---
  softmax_head<<<GH, 256, 0, stream>>>(nqk, natt, GN, 0.125f);               // 1/sqrt(64)
  softmax_head<<<GH, 256, 0, stream>>>(eqk, eatt, GE, 0.17677669529663687f); // 1/sqrt(32)

  // --- Sparse message passing (the dense [H,N,N] matmuls are E-sparse scatters) ---
  zero_f32<<<1024, 256, 0, stream>>>(mp_n, (size_t)GH * GN * GFN + (size_t)GH * GE * GFE);
  {
    const int wn = GE * GH * GFN;     // 1,048,576 fp32 atomics
    scatter_node<<<(wn + 255) / 256, 256, 0, stream>>>(eatt, nv, src_g, dst_g, mp_n);
    const int we = GELG * GH * GFE;   // 2,097,152 fp32 atomics
    scatter_edge<<<(we + 255) / 256, 256, 0, stream>>>(natt, ev, src_lg, dst_lg, dst_g, mp_e);
  }

  // --- Gate by values, mean over heads, concat outputs ---
  const int outN = GN * GFN + GE * GFE;
  finalize<<<(outN + 255) / 256, 256, 0, stream>>>(mp_n, nv, mp_e, ev, (float*)d_out);
}